// GATCONV_26792005993050
// MI455X (gfx1250) — compile-verified
//
#include <hip/hip_runtime.h>
#include <hip/hip_bf16.h>

// ---------------------------------------------------------------------------
// GATv2 (2 conv layers) + 2-layer MLP + edge classifier for gfx1250 (MI455X)
// Dense GEMMs use V_WMMA_F32_16X16X4_F32 (fp32 matrix core path, wave32).
// Graph softmax-aggregation uses f32 global atomics + ordered-uint atomicMax.
// ---------------------------------------------------------------------------

#define HEADS 4
#define DIM   32
#define HC    128     // HEADS*DIM
#define NC    3
#define NEG_SLOPE 0.2f

typedef __attribute__((ext_vector_type(2))) float v2f;
typedef __attribute__((ext_vector_type(8))) float v8f;

// ---- ordered-key encoding for float atomic max over unsigned int -----------
__device__ __forceinline__ unsigned int fkey(float f) {
    unsigned int u = __float_as_uint(f);
    return (u & 0x80000000u) ? ~u : (u | 0x80000000u);
}
__device__ __forceinline__ float fkey_dec(unsigned int k) {
    unsigned int u = (k & 0x80000000u) ? (k ^ 0x80000000u) : ~k;
    return __uint_as_float(u);
}

// ---------------------------------------------------------------------------
// WMMA GEMM: Y[M x NOUT] = X[M x K] @ W[K x NOUT] + bias, optional ReLU.
// One wave (32 threads) per 16x16 output tile; K fully unrolled (mult of 4).
// K and NOUT are compile-time so all W offsets fold into load immediates.
// Out-of-range A rows are clamped (not zeroed): A row m only feeds D row m,
// and those rows are masked at the store, so clamped garbage is harmless.
// ---------------------------------------------------------------------------
template <int K, int NOUT>
__global__ __launch_bounds__(32)
void wmma_gemm_bias(const float* __restrict__ X, const float* __restrict__ W,
                    const float* __restrict__ bias, float* __restrict__ Y,
                    int M, int relu)
{
    constexpr int ntilesN = NOUT / 16;
    const int tile = blockIdx.x;
    const int tm = tile / ntilesN;
    const int tn = tile - tm * ntilesN;
    const int lane = threadIdx.x;          // 0..31 (wave32)

    const int mrow = tm * 16 + (lane & 15);
    const int ncol = tn * 16 + (lane & 15);
    const int kgrp = (lane >> 4) * 2;      // 0 or 2

    const int mclamped = (mrow < M) ? mrow : (M - 1);
    const float* xrow = X + (size_t)mclamped * K + kgrp;  // per-lane A base
    const float* wcol = W + ncol;                         // per-lane B base

    v8f acc = {};
#pragma unroll
    for (int k0 = 0; k0 < K; k0 += 4) {
        // A fragment: lane holds A[m][k0+kgrp], A[m][k0+kgrp+1] (contiguous)
        const v2f a = *(const v2f*)(xrow + k0);
        // B fragment: lane holds W[k0+kgrp][ncol], W[k0+kgrp+1][ncol]
        v2f b;
        b.x = wcol[(k0 + 0) * NOUT + kgrp * NOUT];
        b.y = wcol[(k0 + 1) * NOUT + kgrp * NOUT];
        acc = __builtin_amdgcn_wmma_f32_16x16x4_f32(
                  false, a, false, b, (short)0, acc, false, false);
    }

    // C/D layout: VGPR r -> row = r + 8*(lane>>4), col = lane&15
    const int rbase = tm * 16 + (lane >> 4) * 8;
    const float bv = bias[ncol];
#pragma unroll
    for (int r = 0; r < 8; ++r) {
        const int row = rbase + r;
        if (row < M) {
            float v = acc[r] + bv;
            if (relu) v = fmaxf(v, 0.0f);
            Y[(size_t)row * NOUT + ncol] = v;
        }
    }
}

// ---------------------------------------------------------------------------
// Per-(edge, head) logits: leaky_relu(xl[src]+xr[dst]) . att[h]; atomicMax dst.
// Edges e in [0,E) come from edge_index; e in [E, E+N) are self loops.
// ---------------------------------------------------------------------------
__global__ void edge_logits_kernel(const float* __restrict__ xl,
                                   const float* __restrict__ xr,
                                   const float* __restrict__ att,
                                   const int* __restrict__ srcE,
                                   const int* __restrict__ dstE,
                                   float* __restrict__ logits,
                                   unsigned int* __restrict__ dmax,
                                   int E, int N)
{
    const size_t t = (size_t)blockIdx.x * blockDim.x + threadIdx.x;
    const size_t total = (size_t)(E + N) * HEADS;
    if (t >= total) return;
    const int e = (int)(t >> 2);
    const int h = (int)(t & 3);
    int s, d;
    if (e < E) { s = srcE[e]; d = dstE[e]; } else { s = e - E; d = s; }

    const float4* pl = (const float4*)(xl + (size_t)s * HC + h * DIM);
    const float4* pr = (const float4*)(xr + (size_t)d * HC + h * DIM);
    const float4* pa = (const float4*)(att + h * DIM);

    float acc = 0.0f;
#pragma unroll
    for (int i = 0; i < DIM / 4; ++i) {
        const float4 a = pl[i], b = pr[i], w = pa[i];
        float s0 = a.x + b.x; s0 = (s0 > 0.0f) ? s0 : NEG_SLOPE * s0;
        float s1 = a.y + b.y; s1 = (s1 > 0.0f) ? s1 : NEG_SLOPE * s1;
        float s2 = a.z + b.z; s2 = (s2 > 0.0f) ? s2 : NEG_SLOPE * s2;
        float s3 = a.w + b.w; s3 = (s3 > 0.0f) ? s3 : NEG_SLOPE * s3;
        acc = fmaf(s0, w.x, acc);
        acc = fmaf(s1, w.y, acc);
        acc = fmaf(s2, w.z, acc);
        acc = fmaf(s3, w.w, acc);
    }
    logits[t] = acc;                                   // [e][h]
    atomicMax(dmax + (size_t)d * HEADS + h, fkey(acc));
}

// ---------------------------------------------------------------------------
// a = exp(logit - max[dst]); denom[dst] += a  (unnormalized alpha kept)
// ---------------------------------------------------------------------------
__global__ void edge_exp_kernel(const float* __restrict__ logits,
                                const unsigned int* __restrict__ dmax,
                                const int* __restrict__ dstE,
                                float* __restrict__ alpha,
                                float* __restrict__ denom,
                                int E, int N)
{
    const size_t t = (size_t)blockIdx.x * blockDim.x + threadIdx.x;
    const size_t total = (size_t)(E + N) * HEADS;
    if (t >= total) return;
    const int e = (int)(t >> 2);
    const int h = (int)(t & 3);
    const int d = (e < E) ? dstE[e] : (e - E);
    const float m = fkey_dec(dmax[(size_t)d * HEADS + h]);
    const float a = __expf(logits[t] - m);
    alpha[t] = a;
    atomicAdd(denom + (size_t)d * HEADS + h, a);
}

// ---------------------------------------------------------------------------
// accum[dst][c] += xl[src][c] * alpha/(denom+eps).  128 consecutive threads
// cover one edge's 128 channels -> coalesced atomics on one dst row.
// ---------------------------------------------------------------------------
__global__ void edge_aggregate_kernel(const float* __restrict__ xl,
                                      const float* __restrict__ alpha,
                                      const float* __restrict__ denom,
                                      const int* __restrict__ srcE,
                                      const int* __restrict__ dstE,
                                      float* __restrict__ accum,
                                      int E, int N)
{
    const size_t t = (size_t)blockIdx.x * blockDim.x + threadIdx.x;
    const size_t total = (size_t)(E + N) * HC;
    if (t >= total) return;
    const int e = (int)(t >> 7);
    const int c = (int)(t & (HC - 1));
    const int h = c >> 5;
    int s, d;
    if (e < E) { s = srcE[e]; d = dstE[e]; } else { s = e - E; d = s; }
    const float al = alpha[(size_t)e * HEADS + h] /
                     (denom[(size_t)d * HEADS + h] + 1e-16f);
    atomicAdd(accum + (size_t)d * HC + c, xl[(size_t)s * HC + c] * al);
}

// ---------------------------------------------------------------------------
// out = relu(accum + bias[c])
// ---------------------------------------------------------------------------
__global__ void bias_relu_kernel(const float* __restrict__ accum,
                                 const float* __restrict__ bias,
                                 float* __restrict__ out, int N)
{
    const size_t t = (size_t)blockIdx.x * blockDim.x + threadIdx.x;
    if (t >= (size_t)N * HC) return;
    out[t] = fmaxf(accum[t] + bias[t & (HC - 1)], 0.0f);
}

// ---------------------------------------------------------------------------
// e = concat(h2[src0], h2[dst0]); out = e @ clsW + clsb. One thread per edge.
// ---------------------------------------------------------------------------
__global__ void classify_kernel(const float* __restrict__ h2,
                                const int* __restrict__ src0,
                                const int* __restrict__ dst0,
                                const float* __restrict__ clsW,
                                const float* __restrict__ clsb,
                                float* __restrict__ out,
                                float* __restrict__ efeat, int E)
{
    const int e = blockIdx.x * blockDim.x + threadIdx.x;
    if (e >= E) return;
    const int s = src0[e], d = dst0[e];
    const float4* ps = (const float4*)(h2 + (size_t)s * DIM);
    const float4* pd = (const float4*)(h2 + (size_t)d * DIM);
    float4* pe = (float4*)(efeat + (size_t)e * 2 * DIM);

    float o0 = clsb[0], o1 = clsb[1], o2 = clsb[2];
#pragma unroll
    for (int i = 0; i < DIM / 4; ++i) {
        const float4 v = ps[i];
        pe[i] = v;
        const int kb = i * 4;
        o0 = fmaf(v.x, clsW[(kb + 0) * NC + 0], o0);
        o1 = fmaf(v.x, clsW[(kb + 0) * NC + 1], o1);
        o2 = fmaf(v.x, clsW[(kb + 0) * NC + 2], o2);
        o0 = fmaf(v.y, clsW[(kb + 1) * NC + 0], o0);
        o1 = fmaf(v.y, clsW[(kb + 1) * NC + 1], o1);
        o2 = fmaf(v.y, clsW[(kb + 1) * NC + 2], o2);
        o0 = fmaf(v.z, clsW[(kb + 2) * NC + 0], o0);
        o1 = fmaf(v.z, clsW[(kb + 2) * NC + 1], o1);
        o2 = fmaf(v.z, clsW[(kb + 2) * NC + 2], o2);
        o0 = fmaf(v.w, clsW[(kb + 3) * NC + 0], o0);
        o1 = fmaf(v.w, clsW[(kb + 3) * NC + 1], o1);
        o2 = fmaf(v.w, clsW[(kb + 3) * NC + 2], o2);
    }
#pragma unroll
    for (int i = 0; i < DIM / 4; ++i) {
        const float4 v = pd[i];
        pe[DIM / 4 + i] = v;
        const int kb = DIM + i * 4;
        o0 = fmaf(v.x, clsW[(kb + 0) * NC + 0], o0);
        o1 = fmaf(v.x, clsW[(kb + 0) * NC + 1], o1);
        o2 = fmaf(v.x, clsW[(kb + 0) * NC + 2], o2);
        o0 = fmaf(v.y, clsW[(kb + 1) * NC + 0], o0);
        o1 = fmaf(v.y, clsW[(kb + 1) * NC + 1], o1);
        o2 = fmaf(v.y, clsW[(kb + 1) * NC + 2], o2);
        o0 = fmaf(v.z, clsW[(kb + 2) * NC + 0], o0);
        o1 = fmaf(v.z, clsW[(kb + 2) * NC + 1], o1);
        o2 = fmaf(v.z, clsW[(kb + 2) * NC + 2], o2);
        o0 = fmaf(v.w, clsW[(kb + 3) * NC + 0], o0);
        o1 = fmaf(v.w, clsW[(kb + 3) * NC + 1], o1);
        o2 = fmaf(v.w, clsW[(kb + 3) * NC + 2], o2);
    }
    out[(size_t)e * NC + 0] = o0;
    out[(size_t)e * NC + 1] = o1;
    out[(size_t)e * NC + 2] = o2;
}

// ---------------------------------------------------------------------------
static inline unsigned int cdiv(size_t a, size_t b) {
    return (unsigned int)((a + b - 1) / b);
}

extern "C" void kernel_launch(void* const* d_in, const int* in_sizes, int n_in,
                              void* d_out, int out_size, void* d_ws, size_t ws_size,
                              hipStream_t stream)
{
    // ---- inputs (setup_inputs dict order) ----
    const float* x        = (const float*)d_in[0];
    const int*   eidx     = (const int*)d_in[1];
    const int    N        = in_sizes[2];            // batch has N entries
    const int    E        = in_sizes[1] / 2;
    const int*   srcE     = eidx;
    const int*   dstE     = eidx + E;

    const float* c_Wl[2]  = { (const float*)d_in[3],  (const float*)d_in[9]  };
    const float* c_bl[2]  = { (const float*)d_in[4],  (const float*)d_in[10] };
    const float* c_Wr[2]  = { (const float*)d_in[5],  (const float*)d_in[11] };
    const float* c_br[2]  = { (const float*)d_in[6],  (const float*)d_in[12] };
    const float* c_att[2] = { (const float*)d_in[7],  (const float*)d_in[13] };
    const float* c_b[2]   = { (const float*)d_in[8],  (const float*)d_in[14] };
    const float* l1_W = (const float*)d_in[15];
    const float* l1_b = (const float*)d_in[16];
    const float* l2_W = (const float*)d_in[17];
    const float* l2_b = (const float*)d_in[18];
    const float* clsW = (const float*)d_in[19];
    const float* clsb = (const float*)d_in[20];

    // ---- workspace layout ----
    const size_t E2    = (size_t)E + N;
    float* ws     = (float*)d_ws;
    float* xl     = ws;                      // N*HC
    float* xr     = xl + (size_t)N * HC;     // N*HC
    float* accum  = xr + (size_t)N * HC;     // N*HC
    float* hbuf   = accum + (size_t)N * HC;  // N*HC
    float* logits = hbuf + (size_t)N * HC;   // E2*HEADS
    float* alpha  = logits + E2 * HEADS;     // E2*HEADS
    unsigned int* dmax = (unsigned int*)(alpha + E2 * HEADS);  // N*HEADS
    float* denom  = (float*)(dmax + (size_t)N * HEADS);        // N*HEADS
    float* h1     = denom + (size_t)N * HEADS;                 // N*DIM
    float* h2     = h1 + (size_t)N * DIM;                      // N*DIM

    const int tilesM  = (N + 15) / 16;
    const unsigned gEH  = cdiv(E2 * HEADS, 256);
    const unsigned gEC  = cdiv(E2 * HC, 256);
    const unsigned gNC128 = cdiv((size_t)N * HC, 256);

    const float* lin = x;   // layer input
    for (int l = 0; l < 2; ++l) {
        // zero accumulation state (safe for graph replay)
        hipMemsetAsync(dmax,  0, (size_t)N * HEADS * sizeof(unsigned int), stream);
        hipMemsetAsync(denom, 0, (size_t)N * HEADS * sizeof(float), stream);
        hipMemsetAsync(accum, 0, (size_t)N * HC * sizeof(float), stream);

        // xl = lin @ Wl + bl ; xr = lin @ Wr + br   (K = F = HC = 128)
        wmma_gemm_bias<HC, HC><<<tilesM * (HC / 16), 32, 0, stream>>>(
            lin, c_Wl[l], c_bl[l], xl, N, 0);
        wmma_gemm_bias<HC, HC><<<tilesM * (HC / 16), 32, 0, stream>>>(
            lin, c_Wr[l], c_br[l], xr, N, 0);

        edge_logits_kernel<<<gEH, 256, 0, stream>>>(
            xl, xr, c_att[l], srcE, dstE, logits, dmax, E, N);
        edge_exp_kernel<<<gEH, 256, 0, stream>>>(
            logits, dmax, dstE, alpha, denom, E, N);
        edge_aggregate_kernel<<<gEC, 256, 0, stream>>>(
            xl, alpha, denom, srcE, dstE, accum, E, N);
        bias_relu_kernel<<<gNC128, 256, 0, stream>>>(accum, c_b[l], hbuf, N);

        lin = hbuf;
    }

    // MLP head: h1 = relu(h @ l1_W + b), h2 = relu(h1 @ l2_W + b)
    wmma_gemm_bias<HC, DIM><<<tilesM * (DIM / 16), 32, 0, stream>>>(
        hbuf, l1_W, l1_b, h1, N, 1);
    wmma_gemm_bias<DIM, DIM><<<tilesM * (DIM / 16), 32, 0, stream>>>(
        h1, l2_W, l2_b, h2, N, 1);

    // classifier + edge features (tuple output: [out E*NC | e E*2*DIM])
    float* out_logits = (float*)d_out;
    float* efeat      = out_logits + (size_t)E * NC;
    classify_kernel<<<cdiv((size_t)E, 256), 256, 0, stream>>>(
        h2, srcE, dstE, clsW, clsb, out_logits, efeat, E);

    (void)n_in; (void)out_size; (void)ws_size;
}